// PatchMamba_69922067578912
// MI455X (gfx1250) — compile-verified
//
#include <hip/hip_runtime.h>
#include <math.h>

// ---------------------------------------------------------------------------
// Mamba (HiPPO-LegS) forward for MI455X / gfx1250.
//   D_MODEL=256, N_LAYER=4, D_STATE=16, D_INNER=512, DT_RANK=16, B=8, L=512
// GEMMs use V_WMMA_F32_16X16X4_F32 (keeps f32 reference precision; GEMMs are
// far from the bottleneck).  The selective scan exploits the closed form of
// the LegS A matrix (lower-triangular rank-structure) to turn the 16x16
// matvec per step into a prefix sum (O(16) work / lane / step).
// ---------------------------------------------------------------------------

typedef __attribute__((ext_vector_type(2))) float v2f;
typedef __attribute__((ext_vector_type(8))) float v8f;

#define D_MODEL 256
#define D_STATE 16
#define D_INNER 512
#define DT_RANK 16
#define D_CONV  4
#define B_SZ    8
#define SEQ_L   512
#define ROWS    (B_SZ * SEQ_L)      // 4096
#define XDBL_N  (DT_RANK + 2 * D_STATE)  // 48

__device__ __forceinline__ float softplus_f(float x) {
    return (x > 20.f) ? x : log1pf(__expf(x));
}
__device__ __forceinline__ float silu_f(float x) {
    return x / (1.f + __expf(-x));
}

// ---------------------------------------------------------------------------
// Generic C[M,N] = A[M,K] * W[N,K]^T  (+bias) (+residual) (+softplus)
// one 16x16 output tile per wave32, K-loop in steps of 4 via WMMA f32 16x16x4
// ---------------------------------------------------------------------------
__global__ __launch_bounds__(128) void gemm_wmma_f32_kernel(
    const float* __restrict__ A, int lda,
    const float* __restrict__ W,          // N x K, row-major (ldb == K)
    float* __restrict__ C, int ldc,
    int M, int N, int K,
    const float* __restrict__ bias,       // nullable, length N
    const float* __restrict__ residual,   // nullable, same layout as C
    int act)                              // 0 = none, 1 = softplus
{
    const int wave = threadIdx.x >> 5;
    const int lane = threadIdx.x & 31;
    const int tiles_n = N >> 4;
    const int total_tiles = (M >> 4) * tiles_n;
    const int tile = blockIdx.x * 4 + wave;
    if (tile >= total_tiles) return;      // wave-uniform; EXEC stays all-ones

    const int m_base = (tile / tiles_n) << 4;
    const int n_base = (tile % tiles_n) << 4;

    // A-frag: lanes 0-15 -> rows M=0..15 (K=k+0,k+1), lanes 16-31 -> K=k+2,k+3
    const int half = lane >> 4;           // 0 or 1
    const int kk   = half << 1;           // 0 or 2
    const int mrow = m_base + (lane & 15);
    const int nrow = n_base + (lane & 15);

    const float2* a2 = reinterpret_cast<const float2*>(A + (size_t)mrow * lda + kk);
    const float2* b2 = reinterpret_cast<const float2*>(W + (size_t)nrow * K + kk);

    v8f acc = {};
    for (int k = 0; k < K; k += 4) {
        float2 av = a2[k >> 1];
        float2 bv = b2[k >> 1];
        v2f a, b;
        a.x = av.x; a.y = av.y;
        b.x = bv.x; b.y = bv.y;
        acc = __builtin_amdgcn_wmma_f32_16x16x4_f32(
            /*neg_a=*/false, a, /*neg_b=*/false, b,
            /*c_mod=*/(short)0, acc, /*reuse_a=*/false, /*reuse_b=*/false);
    }

    // C/D layout: lanes 0-15: VGPR r -> M=r; lanes 16-31: VGPR r -> M=r+8
    const int n  = n_base + (lane & 15);
    const int mo = m_base + (half << 3);
    const float bi = bias ? bias[n] : 0.f;
#pragma unroll
    for (int r = 0; r < 8; ++r) {
        float v = acc[r] + bi;
        if (act == 1) v = softplus_f(v);
        const size_t idx = (size_t)(mo + r) * ldc + n;
        if (residual) v += residual[idx];
        C[idx] = v;
    }
}

// ---------------------------------------------------------------------------
// Causal depthwise conv (width 4, left pad 3) + bias + SiLU.
// xc = Y1[..., 0:512] (row stride 1024).  u[b,l,d] out (row stride 512).
// ---------------------------------------------------------------------------
__global__ __launch_bounds__(256) void conv_silu_kernel(
    const float* __restrict__ y1,         // (ROWS, 1024); xc in cols [0,512)
    const float* __restrict__ cw,         // (512, 4)
    const float* __restrict__ cb,         // (512)
    float* __restrict__ u)                // (ROWS, 512)
{
    const int idx = blockIdx.x * 256 + threadIdx.x;   // ROWS*512 threads
    const int d  = idx & (D_INNER - 1);
    const int bl = idx >> 9;
    const int l  = bl & (SEQ_L - 1);
    const int b  = bl >> 9;

    float acc = cb[d];
#pragma unroll
    for (int j = 0; j < D_CONV; ++j) {
        const int ls = l - (D_CONV - 1) + j;
        if (ls >= 0)
            acc += cw[d * D_CONV + j] *
                   y1[((size_t)(b * SEQ_L + ls)) * (2 * D_INNER) + d];
    }
    u[(size_t)bl * D_INNER + d] = silu_f(acc);
}

// ---------------------------------------------------------------------------
// Selective scan.  One block per batch b (512 threads = 16 waves), one lane
// per d_inner channel.  LegS closed form:
//   (s A^T)[n] = -(n+1) s[n] - sq[n] * sum_{m<n} sq[m] s[m],  sq[n]=sqrt(2n+1)
//   B_h[n] = sq[n]
// Fuses the  z = (y + u*D) * silu(res)  epilogue.
// ---------------------------------------------------------------------------
__global__ __launch_bounds__(512) void scan_kernel(
    const float* __restrict__ delta,      // (ROWS, 512)
    const float* __restrict__ u,          // (ROWS, 512)
    const float* __restrict__ xdbl,       // (ROWS, 48); C at cols [32,48)
    const float* __restrict__ y1,         // (ROWS, 1024); res at cols [512,1024)
    const float* __restrict__ Dp,         // (512)
    float* __restrict__ z)                // (ROWS, 512)
{
    const int b = blockIdx.x;
    const int d = threadIdx.x;

    float s[D_STATE];
    float sq[D_STATE];
#pragma unroll
    for (int n = 0; n < D_STATE; ++n) {
        s[n]  = 0.f;
        sq[n] = __fsqrt_rn(2.f * (float)n + 1.f);
    }
    const float dp = Dp[d];

    const float* dptr = delta + (size_t)b * SEQ_L * D_INNER + d;
    const float* uptr = u     + (size_t)b * SEQ_L * D_INNER + d;
    const float* cptr = xdbl  + (size_t)b * SEQ_L * XDBL_N + DT_RANK + D_STATE;
    const float* rptr = y1    + (size_t)b * SEQ_L * (2 * D_INNER) + D_INNER + d;
    float*       zptr = z     + (size_t)b * SEQ_L * D_INNER + d;

    for (int t = 0; t < SEQ_L; ++t) {
        const float dt  = dptr[(size_t)t * D_INNER];
        const float ut  = uptr[(size_t)t * D_INNER];
        const float res = rptr[(size_t)t * (2 * D_INNER)];

        float cvals[D_STATE];                 // wave-uniform address -> s_load
#pragma unroll
        for (int n = 0; n < D_STATE; ++n)
            cvals[n] = cptr[(size_t)t * XDBL_N + n];

        float prefix = 0.f;                   // sum_{m<n} sq[m]*s_old[m]
        float y = 0.f;
#pragma unroll
        for (int n = 0; n < D_STATE; ++n) {
            const float as = -(float)(n + 1) * s[n] - sq[n] * prefix;
            prefix += sq[n] * s[n];           // uses old s[n]
            s[n] += dt * (as + ut * sq[n]);   // B_h[n] = sq[n]
            y += s[n] * cvals[n];             // y uses updated state
        }
        zptr[(size_t)t * D_INNER] = (y + ut * dp) * silu_f(res);
    }
}

// ---------------------------------------------------------------------------
extern "C" void kernel_launch(void* const* d_in, const int* in_sizes, int n_in,
                              void* d_out, int out_size, void* d_ws, size_t ws_size,
                              hipStream_t stream) {
    const float* x_in   = (const float*)d_in[0];   // (8,512,256)
    const float* ipw    = (const float*)d_in[1];   // (4,1024,256)
    const float* cw     = (const float*)d_in[2];   // (4,512,1,4)
    const float* cb     = (const float*)d_in[3];   // (4,512)
    const float* xpw    = (const float*)d_in[4];   // (4,48,512)
    const float* dpw    = (const float*)d_in[5];   // (4,512,16)
    const float* dpb    = (const float*)d_in[6];   // (4,512)
    const float* Dprm   = (const float*)d_in[7];   // (4,512)
    const float* opw    = (const float*)d_in[8];   // (4,256,512)
    float* out = (float*)d_out;

    // workspace layout (floats), 256B-aligned regions
    char* ws = (char*)d_ws;
    auto take = [&](size_t nfloats) {
        float* p = (float*)ws;
        ws += ((nfloats * sizeof(float)) + 255) & ~(size_t)255;
        return p;
    };
    float* bufA  = take((size_t)ROWS * D_MODEL);        // 4 MB
    float* bufB  = take((size_t)ROWS * D_MODEL);        // 4 MB
    float* Y1    = take((size_t)ROWS * 2 * D_INNER);    // 16 MB
    float* U     = take((size_t)ROWS * D_INNER);        // 8 MB
    float* XDBL  = take((size_t)ROWS * XDBL_N);         // 0.75 MB
    float* DELTA = take((size_t)ROWS * D_INNER);        // 8 MB
    float* Z     = take((size_t)ROWS * D_INNER);        // 8 MB
    (void)ws_size; (void)in_sizes; (void)n_in; (void)out_size;

    auto gemm = [&](const float* A, int lda, const float* W, float* C, int ldc,
                    int N, int K, const float* bias, const float* resid, int act) {
        const int tiles = (ROWS / 16) * (N / 16);
        const int blocks = (tiles + 3) / 4;
        gemm_wmma_f32_kernel<<<blocks, 128, 0, stream>>>(
            A, lda, W, C, ldc, ROWS, N, K, bias, resid, act);
    };

    const float* cur = x_in;
    float* outs[4] = { bufA, bufB, bufA, out };

    for (int i = 0; i < 4; ++i) {
        const float* ipw_i = ipw  + (size_t)i * 2 * D_INNER * D_MODEL;
        const float* cw_i  = cw   + (size_t)i * D_INNER * D_CONV;
        const float* cb_i  = cb   + (size_t)i * D_INNER;
        const float* xpw_i = xpw  + (size_t)i * XDBL_N * D_INNER;
        const float* dpw_i = dpw  + (size_t)i * D_INNER * DT_RANK;
        const float* dpb_i = dpb  + (size_t)i * D_INNER;
        const float* Dp_i  = Dprm + (size_t)i * D_INNER;
        const float* opw_i = opw  + (size_t)i * D_MODEL * D_INNER;

        // 1) x_and_res = x @ in_proj_w^T  -> Y1 (ROWS, 1024)
        gemm(cur, D_MODEL, ipw_i, Y1, 2 * D_INNER, 2 * D_INNER, D_MODEL,
             nullptr, nullptr, 0);

        // 2) u = silu(depthwise causal conv(xc) + cb)
        conv_silu_kernel<<<(ROWS * D_INNER) / 256, 256, 0, stream>>>(
            Y1, cw_i, cb_i, U);

        // 3) x_dbl = u @ x_proj_w^T  -> (ROWS, 48)
        gemm(U, D_INNER, xpw_i, XDBL, XDBL_N, XDBL_N, D_INNER,
             nullptr, nullptr, 0);

        // 4) delta = softplus(dt @ dt_proj_w^T + dpb)   (dt = XDBL[:, 0:16])
        gemm(XDBL, XDBL_N, dpw_i, DELTA, D_INNER, D_INNER, DT_RANK,
             dpb_i, nullptr, 1);

        // 5) selective scan (fused +u*D and *silu(res) epilogue) -> Z
        scan_kernel<<<B_SZ, D_INNER, 0, stream>>>(DELTA, U, XDBL, Y1, Dp_i, Z);

        // 6) x = x + Z @ out_proj_w^T  (residual fused into GEMM epilogue)
        gemm(Z, D_INNER, opw_i, outs[i], D_MODEL, D_MODEL, D_INNER,
             nullptr, cur, 0);

        cur = outs[i];
    }
}